// SpectralConvS2_87016037417069
// MI455X (gfx1250) — compile-verified
//
#include <hip/hip_runtime.h>

// ---------------------------------------------------------------------------
// SpectralConvS2 on gfx1250: all FFTs expressed as f32 WMMA GEMMs.
//   out = irfft2( mix_per_lat_mode( rfft2(x) ) ),  residual = x
// All GEMM operands are padded so the hot loop has NO load-side bounds checks:
//   M multiples of 64, K multiples of 16, B-operand widths padded to grid span.
// ---------------------------------------------------------------------------

typedef float v2f __attribute__((ext_vector_type(2)));
typedef float v8f __attribute__((ext_vector_type(8)));

#define BD   2
#define CIN  128
#define COUT 128
#define LATD 360
#define LOND 720
#define QH   361                 // rfft lon modes
#define C2   722                 // re|im stacked (logical)
#define C2P  736                 // C2 padded to mult of 16 (contracted in stage 5)
#define NP768 768                // 722/720 padded to mult of 64 (N-side)
#define ROWS1 92160              // BD*CIN*LATD   (mult of 64)
#define COLS2 92416              // BD*CIN*QH     (= 64*1444)
#define MP768 768                // 720 padded to mult of 64 (M-side, stages 2/4)

static const long long SZ_BUFA = 70975488LL;  // max(92160*768, 768*92416, 360*256*768)
static const long long SZ_BUFB = 70778880LL;  // max(720*92416, 360*256*768, 92160*736)
static const long long W3EL    = 23592960LL;  // 360*256*256
static const long long XEL     = 66355200LL;  // elements of x / out
#define PI_D 3.14159265358979323846

// ======================= f32 WMMA GEMM (padded, unguarded loads) ===========
// C[z] = A[z]*B[z]; row-major. Requirements (guaranteed by caller):
//   gridDim.y*64 rows of A/C are addressable; K % 16 == 0;
//   B/A buffers cover [K x gridDim.x*64] / [gridDim.y*64 x K] at given ld;
//   all base pointers and lds 16-byte aligned. Store masked by gn < N only.
__global__ __launch_bounds__(256)
void gemm_wmma_f32(const float* __restrict__ Ag, const float* __restrict__ Bg,
                   float* __restrict__ Cg,
                   int N, int K, int lda, int ldb, int ldc,
                   long long sA, long long sB, long long sC)
{
    __shared__ float As[64][20];   // row stride 80B: 16B-aligned rows, conflict-free frags
    __shared__ float Bs[16][72];   // row stride 288B: aligned; half-wave banks disjoint

    const float* A = Ag + (long long)blockIdx.z * sA;
    const float* B = Bg + (long long)blockIdx.z * sB;
    float*       C = Cg + (long long)blockIdx.z * sC;

    const int m0 = blockIdx.y * 64;
    const int n0 = blockIdx.x * 64;
    const int t    = threadIdx.x;
    const int wave = t >> 5;
    const int lane = t & 31;
    const int wm   = wave & 3;      // 16-row tile within 64
    const int wn   = wave >> 2;     // 32-col group within 64
    const int lrow = lane & 15;
    const int lhi  = lane >> 4;

    // global staging assignments (float4 per thread per chunk)
    const int ar  = t >> 2;             // A row 0..63
    const int akq = (t & 3) << 2;       // A k offset 0,4,8,12
    const int bkr = t >> 4;             // B k row 0..15
    const int bnq = (t & 15) << 2;      // B n offset 0..60

    const float* Aptr = A + (long long)(m0 + ar) * lda + akq;
    const float* Bptr = B + (long long)bkr * ldb + n0 + bnq;
    const long long bstep = 16LL * ldb;

    v8f acc0 = {0.f,0.f,0.f,0.f,0.f,0.f,0.f,0.f};
    v8f acc1 = acc0;

    for (int k0 = 0; k0 < K; k0 += 16) {
        float4 av = *(const float4*)Aptr;
        float4 bv = *(const float4*)Bptr;
        if (k0 + 16 < K) {                       // uniform branch
            __builtin_prefetch(Aptr + 16, 0, 1); // global_prefetch_b8
            __builtin_prefetch(Bptr + bstep, 0, 1);
        }
        Aptr += 16;
        Bptr += bstep;
        __syncthreads();                         // protect previous frag reads
        *(float4*)&As[ar][akq] = av;
        *(float4*)&Bs[bkr][bnq] = bv;
        __syncthreads();
        #pragma unroll
        for (int ks = 0; ks < 16; ks += 4) {
            // ISA frag layouts: A lane l -> (m=l%16, k=2*(l/16)+j); B mirrored
            v2f a, b0, b1;
            a[0]  = As[wm * 16 + lrow][ks + 2 * lhi + 0];
            a[1]  = As[wm * 16 + lrow][ks + 2 * lhi + 1];
            b0[0] = Bs[ks + 2 * lhi + 0][wn * 32 + lrow];
            b0[1] = Bs[ks + 2 * lhi + 1][wn * 32 + lrow];
            b1[0] = Bs[ks + 2 * lhi + 0][wn * 32 + 16 + lrow];
            b1[1] = Bs[ks + 2 * lhi + 1][wn * 32 + 16 + lrow];
            acc0 = __builtin_amdgcn_wmma_f32_16x16x4_f32(false, a, false, b0,
                                                         (short)0, acc0, false, false);
            acc1 = __builtin_amdgcn_wmma_f32_16x16x4_f32(false, a, false, b1,
                                                         (short)0, acc1, false, false);
        }
    }
    __syncthreads();

    // C/D layout: element e -> (m = e + 8*lhi, n = lrow)
    const int gn0 = n0 + wn * 32 + lrow;
    const int gn1 = gn0 + 16;
    #pragma unroll
    for (int e = 0; e < 8; ++e) {
        int gm = m0 + wm * 16 + lhi * 8 + e;
        if (gn0 < N) C[(long long)gm * ldc + gn0] = acc0[e];
        if (gn1 < N) C[(long long)gm * ldc + gn1] = acc1[e];
    }
}

// ======================= DFT basis construction ============================
// T1[720][768]: lon rfft (ortho). cols [0,361)=re, [361,722)=im, [722,768)=0
__global__ __launch_bounds__(256) void k_init_T1(float* T1)
{
    long long t = (long long)blockIdx.x * 256 + threadIdx.x;
    if (t >= (long long)LOND * NP768) return;
    int n = (int)(t / NP768), c = (int)(t % NP768);
    double v = 0.0;
    if (c < C2) {
        int q = (c < QH) ? c : c - QH;
        double th = 2.0 * PI_D * (double)n * (double)q / (double)LOND;
        double s  = 1.0 / sqrt((double)LOND);
        v = (c < QH) ? cos(th) * s : -sin(th) * s;
    }
    T1[t] = (float)v;
}

// Mlat[768][720]: forward lat FFT block matrix [[C,-S],[S,C]], F=e^{-i th}/sqrt(360)
// rows [720,768) are zero (M-padding).
__global__ __launch_bounds__(256) void k_init_Mlat(float* Mf)
{
    long long t = (long long)blockIdx.x * 256 + threadIdx.x;
    if (t >= (long long)MP768 * LOND) return;
    int r = (int)(t / LOND), c = (int)(t % LOND);
    double v = 0.0;
    if (r < LOND) {
        int p = r % LATD, m = c % LATD;
        int sr = r / LATD, sc = c / LATD;
        double th = 2.0 * PI_D * (double)p * (double)m / (double)LATD;
        double s  = 1.0 / sqrt((double)LATD);
        if (sr == 0 && sc == 0)      v =  cos(th) * s;
        else if (sr == 0)            v =  sin(th) * s;   // -SF
        else if (sc == 0)            v = -sin(th) * s;   //  SF
        else                         v =  cos(th) * s;
    }
    Mf[t] = (float)v;
}

// Minv[768][720]: inverse lat FFT block matrix, G=e^{+i th}/sqrt(360); pad rows 0
__global__ __launch_bounds__(256) void k_init_Minv(float* Mi)
{
    long long t = (long long)blockIdx.x * 256 + threadIdx.x;
    if (t >= (long long)MP768 * LOND) return;
    int r = (int)(t / LOND), c = (int)(t % LOND);
    double v = 0.0;
    if (r < LOND) {
        int p = r % LATD, m = c % LATD;
        int sr = r / LATD, sc = c / LATD;
        double th = 2.0 * PI_D * (double)p * (double)m / (double)LATD;
        double s  = 1.0 / sqrt((double)LATD);
        if (sr == 0 && sc == 0)      v =  cos(th) * s;
        else if (sr == 0)            v = -sin(th) * s;   // -SB
        else if (sc == 0)            v =  sin(th) * s;   //  SB
        else                         v =  cos(th) * s;
    }
    Mi[t] = (float)v;
}

// T5[736][768]: lon irfft (Hermitian doubling; DC/Nyquist imag dropped).
// rows [722,736) = 0 (K-padding, contracted!), cols [720,768) = 0.
__global__ __launch_bounds__(256) void k_init_T5(float* T5)
{
    long long t = (long long)blockIdx.x * 256 + threadIdx.x;
    if (t >= (long long)C2P * NP768) return;
    int c = (int)(t / NP768), n = (int)(t % NP768);
    double v = 0.0;
    if (c < C2 && n < LOND) {
        int q = (c < QH) ? c : c - QH;
        double th = 2.0 * PI_D * (double)q * (double)n / (double)LOND;
        double s  = 1.0 / sqrt((double)LOND);
        if (c < QH) {
            double alpha = (q == 0 || q == LATD) ? 1.0 : 2.0;  // q==360: lon Nyquist
            v = alpha * cos(th) * s;
        } else {
            v = (q == 0 || q == LATD) ? 0.0 : -2.0 * sin(th) * s;
        }
    }
    T5[t] = (float)v;
}

// W3[p][256][256] = [[Wre,-Wim],[Wim,Wre]] from weight[k][i][p][2]
__global__ __launch_bounds__(256) void k_build_W3(const float* __restrict__ w,
                                                  float* __restrict__ W3)
{
    long long t = (long long)blockIdx.x * 256 + threadIdx.x;
    if (t >= W3EL) return;
    int p   = (int)(t >> 16);
    int row = (int)((t >> 8) & 255);
    int col = (int)(t & 255);
    int so = row >> 7, k = row & 127;
    int si = col >> 7, i = col & 127;
    long long base = (((long long)k * CIN + i) * LATD + p) * 2;
    float wre = w[base], wim = w[base + 1];
    W3[t] = (so == 0) ? ((si == 0) ? wre : -wim)
                      : ((si == 0) ? wim :  wre);
}

// ======================= layout permutes ===================================
// Y[(b,i,lat)][(s*361+q), ld 768] -> Y2[(s*360+lat)][(b*128+i)*361+q]
__global__ __launch_bounds__(256) void k_tr1(const float* __restrict__ src, float* __restrict__ dst)
{
    long long t = (long long)blockIdx.x * 256 + threadIdx.x;
    if (t >= 720LL * COLS2) return;
    int r = (int)(t / COLS2), c = (int)(t % COLS2);
    int s = r / LATD, lat = r % LATD;
    int bi = c / QH,  q   = c % QH;
    dst[t] = src[(long long)(bi * LATD + lat) * NP768 + s * QH + q];
}

// U2[(s*360+p)][(b,i,q), ld 92416] -> U3[p][s*128+i][(b*361+q), ld 768] (pad cols 0)
__global__ __launch_bounds__(256) void k_tr2(const float* __restrict__ src, float* __restrict__ dst)
{
    long long t = (long long)blockIdx.x * 256 + threadIdx.x;
    if (t >= 360LL * 256LL * NP768) return;
    int p   = (int)(t / (256LL * NP768));
    int rem = (int)(t % (256LL * NP768));
    int row = rem / NP768, col = rem % NP768;
    float v = 0.0f;
    if (col < C2) {
        int s = row >> 7, i = row & 127;
        int b = col / QH, q = col % QH;
        v = src[(long long)(s * LATD + p) * COLS2 + (b * CIN + i) * QH + q];
    }
    dst[t] = v;
}

// V3[p][so*128+k][(b*361+q), ld 768] -> V4[(so*360+p)][(b*128+k)*361+q]
__global__ __launch_bounds__(256) void k_tr3(const float* __restrict__ src, float* __restrict__ dst)
{
    long long t = (long long)blockIdx.x * 256 + threadIdx.x;
    if (t >= 720LL * COLS2) return;
    int r = (int)(t / COLS2), c = (int)(t % COLS2);
    int so = r / LATD, p = r % LATD;
    int bk = c / QH,   q = c % QH;
    int b = bk >> 7, k = bk & 127;
    dst[t] = src[(long long)p * (256LL * NP768) + (so * COUT + k) * NP768 + b * QH + q];
}

// O2[(s*360+lat)][(b,k,q), ld 92416] -> O5[(b*128+k)*360+lat][(s*361+q), ld 736]
// pad cols [722,736) zeroed (contracted by stage-5 GEMM).
__global__ __launch_bounds__(256) void k_tr4(const float* __restrict__ src, float* __restrict__ dst)
{
    long long t = (long long)blockIdx.x * 256 + threadIdx.x;
    if (t >= (long long)ROWS1 * C2P) return;
    int r = (int)(t / C2P), c = (int)(t % C2P);
    float v = 0.0f;
    if (c < C2) {
        int bk = r / LATD, lat = r % LATD;
        int s  = c / QH,   q   = c % QH;
        v = src[(long long)(s * LATD + lat) * COLS2 + (long long)bk * QH + q];
    }
    dst[t] = v;
}

// residual copy (float4)
__global__ __launch_bounds__(256) void k_copy4(const float4* __restrict__ src,
                                               float4* __restrict__ dst, long long n4)
{
    long long t = (long long)blockIdx.x * 256 + threadIdx.x;
    if (t < n4) dst[t] = src[t];
}

// ======================= orchestration =====================================
extern "C" void kernel_launch(void* const* d_in, const int* in_sizes, int n_in,
                              void* d_out, int out_size, void* d_ws, size_t ws_size,
                              hipStream_t stream)
{
    const float* x = (const float*)d_in[0];
    const float* w = (const float*)d_in[1];
    float* out = (float*)d_out;
    float* ws  = (float*)d_ws;

    float* bufA = ws;                                    // 70,975,488 f
    float* bufB = ws + SZ_BUFA;                          // 70,778,880 f
    float* W3   = ws + SZ_BUFA + SZ_BUFB;                // 23,592,960 f
    float* T1   = W3 + W3EL;                             // 720*768
    float* ML   = T1 + (long long)LOND * NP768;          // 768*720
    float* MI   = ML + (long long)MP768 * LOND;          // 768*720
    float* T5   = MI + (long long)MP768 * LOND;          // 736*768

    const dim3 blk(256);
    auto nblk = [](long long n) { return (int)((n + 255) / 256); };

    // ---- bases (rebuilt every call: deterministic, cheap vs GEMMs) ----
    k_init_T1  <<<nblk((long long)LOND * NP768), blk, 0, stream>>>(T1);
    k_init_Mlat<<<nblk((long long)MP768 * LOND), blk, 0, stream>>>(ML);
    k_init_Minv<<<nblk((long long)MP768 * LOND), blk, 0, stream>>>(MI);
    k_init_T5  <<<nblk((long long)C2P * NP768),  blk, 0, stream>>>(T5);
    k_build_W3 <<<nblk(W3EL),                    blk, 0, stream>>>(w, W3);

    // 1) lon rfft: Y[92160x768] = x[92160x720] * T1[720x768]      -> bufA
    gemm_wmma_f32<<<dim3(NP768/64, ROWS1/64, 1), blk, 0, stream>>>(
        x, T1, bufA, NP768, LOND, LOND, NP768, NP768, 0, 0, 0);

    // permute -> Y2[720 x 92416]                                   -> bufB
    k_tr1<<<nblk(720LL * COLS2), blk, 0, stream>>>(bufA, bufB);

    // 2) lat FFT: U2[768x92416] = Mlat[768x720] * Y2[720x92416]    -> bufA
    gemm_wmma_f32<<<dim3(COLS2/64, MP768/64, 1), blk, 0, stream>>>(
        ML, bufB, bufA, COLS2, LOND, LOND, COLS2, COLS2, 0, 0, 0);

    // permute -> U3[360][256][768]                                 -> bufB
    k_tr2<<<nblk(360LL * 256 * NP768), blk, 0, stream>>>(bufA, bufB);

    // 3) per-mode mixing: V3[p] = W3[p][256x256] * U3[p][256x768]  -> bufA
    gemm_wmma_f32<<<dim3(NP768/64, 256/64, LATD), blk, 0, stream>>>(
        W3, bufB, bufA, NP768, 256, 256, NP768, NP768,
        65536LL, 256LL * NP768, 256LL * NP768);

    // permute -> V4[720 x 92416]                                   -> bufB
    k_tr3<<<nblk(720LL * COLS2), blk, 0, stream>>>(bufA, bufB);

    // 4) lat iFFT: O2[768x92416] = Minv[768x720] * V4[720x92416]   -> bufA
    gemm_wmma_f32<<<dim3(COLS2/64, MP768/64, 1), blk, 0, stream>>>(
        MI, bufB, bufA, COLS2, LOND, LOND, COLS2, COLS2, 0, 0, 0);

    // permute -> O5[92160 x 736] (pad cols zeroed)                 -> bufB
    k_tr4<<<nblk((long long)ROWS1 * C2P), blk, 0, stream>>>(bufA, bufB);

    // 5) lon irfft: out[92160x720] = O5[92160x736] * T5[736x768]   -> d_out
    gemm_wmma_f32<<<dim3(NP768/64, ROWS1/64, 1), blk, 0, stream>>>(
        bufB, T5, out, LOND, C2P, C2P, NP768, LOND, 0, 0, 0);

    // residual = x                                                 -> d_out[XEL..)
    k_copy4<<<nblk(XEL / 4), blk, 0, stream>>>(
        (const float4*)x, (float4*)(out + XEL), XEL / 4);
}